// GATv2Block_15848429322414
// MI455X (gfx1250) — compile-verified
//
#include <hip/hip_runtime.h>
#include <math.h>

#define NN   50000
#define EE   800000
#define DDIM 128
#define HH   8
#define CCH  16
#define EDD  32
#define HIDD 512
#define GG   64
#define LLAY 2
#define NEGS 0.2f
#define EPSF 1e-5f

typedef float v2f __attribute__((ext_vector_type(2)));
typedef float v8f __attribute__((ext_vector_type(8)));

// ---- CDNA5 WMMA: D = A(16x4 f32) * B(4x16 f32) + C(16x16 f32), wave32 ----
__device__ __forceinline__ v8f wmma4(v2f a, v2f b, v8f c) {
  return __builtin_amdgcn_wmma_f32_16x16x4_f32(false, a, false, b, (short)0, c,
                                               false, false);
}

// -------------------- small utility kernels --------------------
__global__ void fill_k(float* __restrict__ p, float v, int n) {
  int i = blockIdx.x * blockDim.x + threadIdx.x;
  if (i < n) p[i] = v;
}

__global__ void count_nodes(const int* __restrict__ nb, float* __restrict__ cnt) {
  int i = blockIdx.x * blockDim.x + threadIdx.x;
  if (i < NN) atomicAdd(&cnt[nb[i]], 1.0f);
}

__device__ __forceinline__ void atomicMaxF(float* addr, float val) {
  if (val >= 0.0f)
    atomicMax((int*)addr, __float_as_int(val));
  else
    atomicMin((unsigned int*)addr, __float_as_uint(val));
}

// -------------------- fused node transform: XL = x@Wl+bl, XR = x@Wr+br,
// ACC = x@Wres + bias (residual+bias pre-seeded into attention accumulator) ---
__global__ void node_xform(const float* __restrict__ x,
                           const float* __restrict__ Wl, const float* __restrict__ bl,
                           const float* __restrict__ Wr, const float* __restrict__ br,
                           const float* __restrict__ Ws, const float* __restrict__ bs,
                           float* __restrict__ XL, float* __restrict__ XR,
                           float* __restrict__ ACC) {
  int lane = threadIdx.x & 31;
  int wave = blockIdx.x * 8 + (threadIdx.x >> 5);
  if (wave >= (NN / 16) * (DDIM / 16)) return;
  int ct   = wave & 7;              // 8 column tiles of 16
  int row0 = (wave >> 3) << 4;
  int lo = lane & 15, hi = lane >> 4;
  int col = ct * 16 + lo;
  const float* arow = x + (size_t)(row0 + lo) * DDIM;
  v8f cl = {}; v8f cr = {}; v8f cs = {};
  for (int k = 0; k < DDIM; k += 4) {
    int ka = k + 2 * hi;
    v2f a;  a.x  = arow[ka];                      a.y  = arow[ka + 1];
    v2f b0; b0.x = Wl[(size_t)ka * DDIM + col];   b0.y = Wl[(size_t)(ka + 1) * DDIM + col];
    v2f b1; b1.x = Wr[(size_t)ka * DDIM + col];   b1.y = Wr[(size_t)(ka + 1) * DDIM + col];
    v2f b2; b2.x = Ws[(size_t)ka * DDIM + col];   b2.y = Ws[(size_t)(ka + 1) * DDIM + col];
    cl = wmma4(a, b0, cl);
    cr = wmma4(a, b1, cr);
    cs = wmma4(a, b2, cs);
  }
#pragma unroll
  for (int i = 0; i < 8; ++i) {
    int r = row0 + i + 8 * hi;
    size_t idx = (size_t)r * DDIM + col;
    XL[idx]  = cl[i] + bl[col];
    XR[idx]  = cr[i] + br[col];
    ACC[idx] = cs[i] + bs[col];
  }
}

// -------------------- H1 = ACC @ W1 + b1  (K=128, Dout=512) --------------------
__global__ void gemm_h1(const float* __restrict__ A, const float* __restrict__ W,
                        const float* __restrict__ b, float* __restrict__ O) {
  int lane = threadIdx.x & 31;
  int wave = blockIdx.x * 8 + (threadIdx.x >> 5);
  if (wave >= (NN / 16) * (HIDD / 16)) return;
  int ct   = wave & 31;             // 32 column tiles
  int row0 = (wave >> 5) << 4;
  int lo = lane & 15, hi = lane >> 4;
  int col = ct * 16 + lo;
  const float* arow = A + (size_t)(row0 + lo) * DDIM;
  v8f c = {};
  for (int k = 0; k < DDIM; k += 4) {
    int ka = k + 2 * hi;
    v2f a;  a.x  = arow[ka];                      a.y  = arow[ka + 1];
    v2f bb; bb.x = W[(size_t)ka * HIDD + col];    bb.y = W[(size_t)(ka + 1) * HIDD + col];
    c = wmma4(a, bb, c);
  }
#pragma unroll
  for (int i = 0; i < 8; ++i) {
    int r = row0 + i + 8 * hi;
    O[(size_t)r * HIDD + col] = c[i] + b[col];
  }
}

// ------- XOUT = XIN + relu(H1*scale+shift) @ W2 + b2  (K=512, Dout=128) -------
__global__ void gemm_mlp2(const float* __restrict__ H1b, const float* __restrict__ sc,
                          const float* __restrict__ sh, const float* __restrict__ W,
                          const float* __restrict__ b, const float* __restrict__ XIN,
                          float* __restrict__ XOUT) {
  int lane = threadIdx.x & 31;
  int wave = blockIdx.x * 8 + (threadIdx.x >> 5);
  if (wave >= (NN / 16) * (DDIM / 16)) return;
  int ct   = wave & 7;
  int row0 = (wave >> 3) << 4;
  int lo = lane & 15, hi = lane >> 4;
  int col = ct * 16 + lo;
  const float* arow = H1b + (size_t)(row0 + lo) * HIDD;
  v8f c = {};
  for (int k = 0; k < HIDD; k += 4) {
    int ka = k + 2 * hi;
    float a0 = fmaf(arow[ka],     sc[ka],     sh[ka]);
    float a1 = fmaf(arow[ka + 1], sc[ka + 1], sh[ka + 1]);
    v2f a;  a.x = fmaxf(a0, 0.0f);  a.y = fmaxf(a1, 0.0f);   // BN-affine + ReLU fused
    v2f bb; bb.x = W[(size_t)ka * DDIM + col]; bb.y = W[(size_t)(ka + 1) * DDIM + col];
    c = wmma4(a, bb, c);
  }
#pragma unroll
  for (int i = 0; i < 8; ++i) {
    int r = row0 + i + 8 * hi;
    size_t idx = (size_t)r * DDIM + col;
    XOUT[idx] = XIN[idx] + c[i] + b[col];
  }
}

// ---- attention logits: alpha[e,h] = sum_c lrelu(xl[src]+xr[dst]+ (ea@We))*att
//      edge GEMM (E x 32 x 128) fused here via LDS; also atomic segment-max ----
__global__ void edge_alpha(const float* __restrict__ XL, const float* __restrict__ XR,
                           const int* __restrict__ ei, const float* __restrict__ ea,
                           const float* __restrict__ We, const float* __restrict__ att,
                           float* __restrict__ ALPHA, float* __restrict__ AMAX) {
  __shared__ float sWe[EDD * DDIM];   // 16 KB
  __shared__ float sEa[32 * EDD];     // 4 KB: 32 edges per block
  __shared__ float sAtt[DDIM];
  int t = threadIdx.x;                // 256 threads = 32 edges x 8 heads
  for (int i = t; i < EDD * DDIM; i += 256) sWe[i] = We[i];
  if (t < DDIM) sAtt[t] = att[t];
  int e0 = blockIdx.x * 32;
  for (int i = t; i < 32 * EDD; i += 256) sEa[i] = ea[(size_t)e0 * EDD + i];
  __syncthreads();

  int el = t >> 3, h = t & 7;
  int e = e0 + el;
  int src = ei[e], dst = ei[EE + e];
  const float* xls = XL + (size_t)src * DDIM + h * CCH;
  const float* xrd = XR + (size_t)dst * DDIM + h * CCH;
  const float* eaL = &sEa[el * EDD];
  float acc = 0.0f;
#pragma unroll
  for (int c = 0; c < CCH; ++c) {
    int ch = h * CCH + c;
    float ef = 0.0f;
#pragma unroll
    for (int k = 0; k < EDD; ++k) ef = fmaf(eaL[k], sWe[k * DDIM + ch], ef);
    float mv = xls[c] + xrd[c] + ef;
    mv = mv > 0.0f ? mv : NEGS * mv;              // leaky_relu
    acc = fmaf(mv, sAtt[ch], acc);
  }
  ALPHA[(size_t)e * HH + h] = acc;
  atomicMaxF(&AMAX[(size_t)dst * HH + h], acc);
}

// ---- ex = exp(alpha - amax[dst]); den[dst] += ex (segment softmax denom) ----
__global__ void edge_exden(const int* __restrict__ ei, float* __restrict__ ALPHA,
                           const float* __restrict__ AMAX, float* __restrict__ DEN) {
  size_t i = (size_t)blockIdx.x * 256 + threadIdx.x;   // E*H total, exact
  int e = (int)(i >> 3), h = (int)(i & 7);
  int dst = ei[EE + e];
  float ex = __expf(ALPHA[i] - AMAX[(size_t)dst * HH + h]);
  ALPHA[i] = ex;
  atomicAdd(&DEN[(size_t)dst * HH + h], ex);
}

// ---- ACC[dst] += (ex/den[dst]) * XL[src]  (weighted scatter) ----
__global__ void edge_scatter(const int* __restrict__ ei, const float* __restrict__ ALPHA,
                             const float* __restrict__ DEN, const float* __restrict__ XL,
                             float* __restrict__ ACC) {
  int t = threadIdx.x;                // 256 = 2 edges x 128 channels
  int el = t >> 7, ch = t & 127;
  int e = blockIdx.x * 2 + el;
  int src = ei[e], dst = ei[EE + e];
  int h = ch >> 4;
  float a = ALPHA[(size_t)e * HH + h] / (DEN[(size_t)dst * HH + h] + 1e-16f);
  atomicAdd(&ACC[(size_t)dst * DDIM + ch], a * XL[(size_t)src * DDIM + ch]);
}

// -------------------- BatchNorm stats / finalize --------------------
__global__ void bn_stats(const float* __restrict__ Hb, float* __restrict__ s,
                         float* __restrict__ q) {
  int c = threadIdx.x;                // 512 columns
  float sum = 0.0f, sq = 0.0f;
  for (int r = blockIdx.x; r < NN; r += gridDim.x) {
    float v = Hb[(size_t)r * HIDD + c];
    sum += v; sq = fmaf(v, v, sq);
  }
  atomicAdd(&s[c], sum);
  atomicAdd(&q[c], sq);
}

__global__ void bn_finalize(const float* __restrict__ s, const float* __restrict__ q,
                            const float* __restrict__ g, const float* __restrict__ b,
                            float* __restrict__ sc, float* __restrict__ sh) {
  int c = threadIdx.x;
  float mu  = s[c] / (float)NN;
  float var = fmaxf(q[c] / (float)NN - mu * mu, 0.0f);   // biased, matches torch BN
  float k = g[c] * rsqrtf(var + EPSF);
  sc[c] = k;
  sh[c] = b[c] - mu * k;
}

// -------------------- graph-mode LayerNorm --------------------
__global__ void ln_node_stats(const float* __restrict__ x, const int* __restrict__ nb,
                              float* __restrict__ gsum, float* __restrict__ gsq) {
  __shared__ float ss[DDIM];
  __shared__ float sq[DDIM];
  int c = threadIdx.x;
  float v = x[(size_t)blockIdx.x * DDIM + c];
  ss[c] = v; sq[c] = v * v;
  __syncthreads();
  for (int st = 64; st > 0; st >>= 1) {
    if (c < st) { ss[c] += ss[c + st]; sq[c] += sq[c + st]; }
    __syncthreads();
  }
  if (c == 0) {
    int g = nb[blockIdx.x];
    atomicAdd(&gsum[g], ss[0]);
    atomicAdd(&gsq[g], sq[0]);
  }
}

__global__ void ln_finalize(const float* __restrict__ gsum, const float* __restrict__ gsq,
                            const float* __restrict__ cnt, float* __restrict__ gm,
                            float* __restrict__ gr) {
  int g = threadIdx.x;                // 64 graphs
  float denom = fmaxf(cnt[g], 1.0f) * (float)DDIM;
  float m = gsum[g] / denom;
  float v = fmaxf(gsq[g] / denom - m * m, 0.0f);
  gm[g] = m;
  gr[g] = rsqrtf(v + EPSF);
}

__global__ void ln_apply(float* __restrict__ x, const int* __restrict__ nb,
                         const float* __restrict__ gm, const float* __restrict__ gr,
                         const float* __restrict__ lg, const float* __restrict__ lb) {
  size_t i = (size_t)blockIdx.x * 256 + threadIdx.x;    // N*D exact multiple
  int n = (int)(i >> 7), c = (int)(i & 127);
  int g = nb[n];
  x[i] = (x[i] - gm[g]) * gr[g] * lg[c] + lb[c];
}

// ==================== host launcher ====================
extern "C" void kernel_launch(void* const* d_in, const int* in_sizes, int n_in,
                              void* d_out, int out_size, void* d_ws, size_t ws_size,
                              hipStream_t stream) {
  (void)in_sizes; (void)n_in; (void)out_size; (void)ws_size;
  const float* x_in = (const float*)d_in[0];
  const int*   nb   = (const int*)d_in[1];
  const int*   ei   = (const int*)d_in[2];
  const float* ea   = (const float*)d_in[3];
  const float* Wl   = (const float*)d_in[4];
  const float* bl   = (const float*)d_in[5];
  const float* Wr   = (const float*)d_in[6];
  const float* br   = (const float*)d_in[7];
  const float* We   = (const float*)d_in[8];
  const float* att  = (const float*)d_in[9];
  const float* bias = (const float*)d_in[10];
  const float* Wres = (const float*)d_in[11];
  const float* W1   = (const float*)d_in[12];
  const float* b1   = (const float*)d_in[13];
  const float* bng  = (const float*)d_in[14];
  const float* bnb  = (const float*)d_in[15];
  const float* W2   = (const float*)d_in[16];
  const float* b2   = (const float*)d_in[17];
  const float* lng  = (const float*)d_in[18];
  const float* lnb  = (const float*)d_in[19];

  float* xout = (float*)d_out;           // running x buffer (fully overwritten)
  float* p = (float*)d_ws;
  float* XL    = p; p += (size_t)NN * DDIM;
  float* XR    = p; p += (size_t)NN * DDIM;
  float* ACC   = p; p += (size_t)NN * DDIM;
  float* ALPHA = p; p += (size_t)EE * HH;
  float* AMAX  = p; p += (size_t)NN * HH;
  float* DEN   = p; p += (size_t)NN * HH;
  float* H1    = p; p += (size_t)NN * HIDD;
  float* BNS   = p; p += HIDD;
  float* BNQ   = p; p += HIDD;
  float* BSC   = p; p += HIDD;
  float* BSH   = p; p += HIDD;
  float* GSUM  = p; p += GG;
  float* GSQ   = p; p += GG;
  float* GM    = p; p += GG;
  float* GR    = p; p += GG;
  float* CNT   = p; p += GG;

  fill_k<<<1, 64, 0, stream>>>(CNT, 0.0f, GG);
  count_nodes<<<(NN + 255) / 256, 256, 0, stream>>>(nb, CNT);

  for (int l = 0; l < LLAY; ++l) {
    const float* xcur = (l == 0) ? x_in : xout;
    node_xform<<<(NN / 16) * (DDIM / 16) / 8, 256, 0, stream>>>(
        xcur, Wl + (size_t)l * DDIM * DDIM, bl + l * DDIM,
        Wr + (size_t)l * DDIM * DDIM, br + l * DDIM,
        Wres + (size_t)l * DDIM * DDIM, bias + l * DDIM, XL, XR, ACC);

    fill_k<<<(NN * HH + 255) / 256, 256, 0, stream>>>(AMAX, -INFINITY, NN * HH);
    fill_k<<<(NN * HH + 255) / 256, 256, 0, stream>>>(DEN, 0.0f, NN * HH);
    edge_alpha<<<EE / 32, 256, 0, stream>>>(XL, XR, ei, ea,
        We + (size_t)l * EDD * DDIM, att + l * HH * CCH, ALPHA, AMAX);
    edge_exden<<<(EE * HH) / 256, 256, 0, stream>>>(ei, ALPHA, AMAX, DEN);
    edge_scatter<<<EE / 2, 256, 0, stream>>>(ei, ALPHA, DEN, XL, ACC);

    gemm_h1<<<(NN / 16) * (HIDD / 16) / 8, 256, 0, stream>>>(
        ACC, W1 + (size_t)l * DDIM * HIDD, b1 + l * HIDD, H1);
    fill_k<<<(2 * HIDD + 255) / 256, 256, 0, stream>>>(BNS, 0.0f, 2 * HIDD);
    bn_stats<<<128, HIDD, 0, stream>>>(H1, BNS, BNQ);
    bn_finalize<<<1, HIDD, 0, stream>>>(BNS, BNQ, bng + l * HIDD, bnb + l * HIDD, BSC, BSH);
    gemm_mlp2<<<(NN / 16) * (DDIM / 16) / 8, 256, 0, stream>>>(
        H1, BSC, BSH, W2 + (size_t)l * HIDD * DDIM, b2 + l * DDIM, ACC, xout);

    fill_k<<<1, 128, 0, stream>>>(GSUM, 0.0f, 2 * GG);   // GSUM,GSQ contiguous
    ln_node_stats<<<NN, DDIM, 0, stream>>>(xout, nb, GSUM, GSQ);
    ln_finalize<<<1, GG, 0, stream>>>(GSUM, GSQ, CNT, GM, GR);
    ln_apply<<<(NN * DDIM) / 256, 256, 0, stream>>>(xout, nb, GM, GR,
        lng + l * DDIM, lnb + l * DDIM);
  }
}